// SimpleGNN_69458211111579
// MI455X (gfx1250) — compile-verified
//
#include <hip/hip_runtime.h>
#include <hip/hip_bf16.h>

typedef __attribute__((ext_vector_type(2))) float v2f;
typedef __attribute__((ext_vector_type(8))) float v8f;

#define WPAD 68   // padded column stride for transposed W in LDS (bank-conflict-free)

// ---------------------------------------------------------------------------
// Degree / normalization
// ---------------------------------------------------------------------------
__global__ __launch_bounds__(256) void k_fill1(float* __restrict__ p, int n) {
    int i = blockIdx.x * 256 + threadIdx.x;
    if (i < n) p[i] = 1.0f;                       // self-loop contributes 1
}

__global__ __launch_bounds__(256) void k_deg(const int* __restrict__ dst,
                                             float* __restrict__ deg, int nE) {
    int i = blockIdx.x * 256 + threadIdx.x;
    if (i < nE) atomicAdd(&deg[dst[i]], 1.0f);
}

__global__ __launch_bounds__(256) void k_rsqrt(float* __restrict__ d, int n) {
    int i = blockIdx.x * 256 + threadIdx.x;
    if (i < n) d[i] = rsqrtf(fmaxf(d[i], 1.0f));
}

__global__ __launch_bounds__(256) void k_zero(float* __restrict__ p, long n) {
    long i = (long)blockIdx.x * 256 + threadIdx.x;
    if (i < n) p[i] = 0.0f;
}

// ---------------------------------------------------------------------------
// Dense GEMM: OUT[nrows x 64] = X[nrows x 64] @ W[64 x 64]
// Full-precision V_WMMA_F32_16X16X4_F32, one 16x64 stripe per wave.
// W staged transposed+padded in LDS: B fragment = one aligned ds_load_b64.
// ---------------------------------------------------------------------------
__global__ __launch_bounds__(256) void k_gemm64(const float* __restrict__ X,
                                                const float* __restrict__ W,
                                                float* __restrict__ OUT,
                                                int nrows) {
    __shared__ float sWt[64 * WPAD];               // sWt[col*WPAD + k] = W[k][col]
    for (int i = threadIdx.x; i < 64 * 64; i += 256) {
        int k = i >> 6, col = i & 63;
        sWt[col * WPAD + k] = W[i];
    }
    __syncthreads();

    const int  wave = threadIdx.x >> 5;
    const int  lane = threadIdx.x & 31;
    const long tile = (long)blockIdx.x * 8 + wave;     // 16-row tile index
    if (tile * 16 >= nrows) return;                    // wave-uniform guard

    const long row0 = tile * 16;
    const int  m    = lane & 15;                       // M (and B-col) within tile
    const int  kh   = (lane >> 4) << 1;                // K sub-offset: 0 or 2
    const float* xrow = X + (row0 + m) * 64;

    v8f acc[4] = {};                                   // 4 col-tiles of 16
#pragma unroll
    for (int k0 = 0; k0 < 64; k0 += 4) {
        v2f a = *(const v2f*)(xrow + k0 + kh);         // A: 16x4 fp32 frag
#pragma unroll
        for (int n = 0; n < 4; ++n) {
            // B: 4x16 fp32 frag; contiguous pair in transposed LDS image
            v2f b = *(const v2f*)(sWt + (n * 16 + m) * WPAD + k0 + kh);
            acc[n] = __builtin_amdgcn_wmma_f32_16x16x4_f32(
                false, a, false, b, (short)0, acc[n], false, false);
        }
    }

    const int mhi = (lane >> 4) << 3;                  // 0 or 8
#pragma unroll
    for (int n = 0; n < 4; ++n)
#pragma unroll
        for (int v = 0; v < 8; ++v)
            OUT[(row0 + mhi + v) * 64 + n * 16 + m] = acc[n][v];
}

// ---------------------------------------------------------------------------
// Edge aggregation: out[dst] += dinv[src]*dinv[dst] * h[src]
// One wave per edge, each lane handles 2 features (float2 gather, L2-resident)
// ---------------------------------------------------------------------------
__global__ __launch_bounds__(256) void k_agg(const float* __restrict__ h,
                                             const float* __restrict__ dinv,
                                             const int* __restrict__ src,
                                             const int* __restrict__ dst,
                                             float* __restrict__ out, int nE) {
    long tid = (long)blockIdx.x * 256 + threadIdx.x;
    long e   = tid >> 5;
    if (e >= nE) return;
    int   f    = (int)(tid & 31) << 1;
    int   s    = src[e];
    int   d    = dst[e];
    float norm = dinv[s] * dinv[d];
    v2f hv = *(const v2f*)(h + (long)s * 64 + f);
    atomicAdd(out + (long)d * 64 + f,     norm * hv.x);
    atomicAdd(out + (long)d * 64 + f + 1, norm * hv.y);
}

// ---------------------------------------------------------------------------
// Epilogue: agg += dinv^2 * h (self-loop) + bias, optional ReLU (in place)
// ---------------------------------------------------------------------------
__global__ __launch_bounds__(256) void k_epi(float* __restrict__ agg,
                                             const float* __restrict__ h,
                                             const float* __restrict__ dinv,
                                             const float* __restrict__ bias,
                                             int relu, long nTot) {
    long t = (long)blockIdx.x * 256 + threadIdx.x;
    if (t >= nTot) return;
    long node = t >> 6;
    int  f    = (int)(t & 63);
    float di  = dinv[node];
    float v   = agg[t] + di * di * h[t] + bias[f];
    if (relu) v = fmaxf(v, 0.0f);
    agg[t] = v;
}

// ---------------------------------------------------------------------------
// Head: out[n,c] = h[n,:] @ Wlin[:,c] + blin[c]   (64 x 2, too skinny for WMMA)
// ---------------------------------------------------------------------------
__global__ __launch_bounds__(256) void k_final(const float* __restrict__ h,
                                               const float* __restrict__ Wl,
                                               const float* __restrict__ bl,
                                               float* __restrict__ out, int n) {
    int t = blockIdx.x * 256 + threadIdx.x;
    if (t >= n * 2) return;
    int node = t >> 1;
    int c    = t & 1;
    const float* hr = h + (long)node * 64;
    float acc = bl[c];
#pragma unroll
    for (int k = 0; k < 64; ++k) acc += hr[k] * Wl[k * 2 + c];
    out[t] = acc;
}

// ---------------------------------------------------------------------------
extern "C" void kernel_launch(void* const* d_in, const int* in_sizes, int n_in,
                              void* d_out, int out_size, void* d_ws, size_t ws_size,
                              hipStream_t stream) {
    const float* x  = (const float*)d_in[0];
    const int*   ei = (const int*)d_in[1];   // [2,E] int32 (JAX x64 off)
    // d_in[2] = batch (unused)
    const float* W1 = (const float*)d_in[3];
    const float* b1 = (const float*)d_in[4];
    const float* W2 = (const float*)d_in[5];
    const float* b2 = (const float*)d_in[6];
    const float* Wl = (const float*)d_in[7];
    const float* bl = (const float*)d_in[8];

    const int N = in_sizes[0] / 64;
    const int E = in_sizes[1] / 2;
    const int* src = ei;
    const int* dst = ei + E;

    // workspace layout: dinv[N] | bufH[N*64] | bufA[N*64]  (~52 MB)
    float* ws   = (float*)d_ws;
    size_t nAl  = ((size_t)N + 255) & ~(size_t)255;
    float* dinv = ws;
    float* bufH = ws + nAl;
    float* bufA = bufH + (size_t)N * 64;

    const long NF    = (long)N * 64;
    const int  gN    = (N + 255) / 256;
    const int  gE    = (E + 255) / 256;
    const int  gNF   = (int)((NF + 255) / 256);
    const int  gGemm = (N / 16 + 7) / 8;
    const int  gAgg  = (int)(((long)E * 32 + 255) / 256);
    dim3 blk(256);

    // symmetric normalization dinv = rsqrt(1 + indegree)
    k_fill1<<<gN, blk, 0, stream>>>(dinv, N);
    k_deg  <<<gE, blk, 0, stream>>>(dst, dinv, E);
    k_rsqrt<<<gN, blk, 0, stream>>>(dinv, N);

    // layer 1: h1 = x@W1 ; agg ; + self-loop + b1 ; ReLU
    k_gemm64<<<gGemm, blk, 0, stream>>>(x, W1, bufH, N);
    k_zero  <<<gNF,   blk, 0, stream>>>(bufA, NF);
    k_agg   <<<gAgg,  blk, 0, stream>>>(bufH, dinv, src, dst, bufA, E);
    k_epi   <<<gNF,   blk, 0, stream>>>(bufA, bufH, dinv, b1, 1, NF);

    // layer 2: h2 = relu1@W2 ; agg ; + self-loop + b2
    k_gemm64<<<gGemm, blk, 0, stream>>>(bufA, W2, bufH, N);
    k_zero  <<<gNF,   blk, 0, stream>>>(bufA, NF);
    k_agg   <<<gAgg,  blk, 0, stream>>>(bufH, dinv, src, dst, bufA, E);
    k_epi   <<<gNF,   blk, 0, stream>>>(bufA, bufH, dinv, b2, 0, NF);

    // linear head
    k_final<<<(N * 2 + 255) / 256, blk, 0, stream>>>(bufA, Wl, bl, (float*)d_out, N);
}